// TopKBalancedNoisyGate_28819230556397
// MI455X (gfx1250) — compile-verified
//
#include <hip/hip_runtime.h>
#include <hip/hip_bf16.h>
#include <math.h>

typedef __attribute__((ext_vector_type(2))) float v2f;
typedef __attribute__((ext_vector_type(8))) float v8f;

#define N_ROWS   8192
#define DIM      4096
#define NEXP     16
#define KSEL     4
#define NOISE_EPS 0.01f
#define LOSS_COEF 0.01f
#define NBLOCKS  (N_ROWS / 16)      // 512 row tiles
#define NWAVES   8                  // waves per block (wave32)
#define NTHREADS (NWAVES * 32)      // 256
#define KSLICE   (DIM / NWAVES)     // 512 K per wave

// ---------------------------------------------------------------------------
// Fused gate kernel: one 16-row tile per block.
//   Phase 1: 8 waves each accumulate a 16x16 f32 tile for both GEMMs over
//            their K-slice using V_WMMA_F32_16X16X4_F32. x is streamed with
//            non-temporal loads (single-use, 128MB); weights stay cached.
//   Phase 2: deterministic LDS reduction across waves -> (row,col) layout.
//   Phase 3: 16 threads do the noisy-topk epilogue, write outputs + partials.
// ---------------------------------------------------------------------------
__global__ __launch_bounds__(NTHREADS) void gate_fused_kernel(
    const float* __restrict__ x,      // (8192, 4096)
    const float* __restrict__ wg,     // (4096, 16)
    const float* __restrict__ wn,     // (4096, 16)
    const float* __restrict__ noise,  // (8192, 16)
    float* __restrict__ out,          // 65537 floats
    float* __restrict__ ws)           // >= 512*32 floats
{
  __shared__ float s_part[NWAVES * 2 * 32 * 8];  // [wave][mat][lane][vgpr] 16KB
  __shared__ float sG[256];                 // logits_gate, (row,col)
  __shared__ float sZ[256];                 // pre-softplus, then noise_control
  __shared__ float sLN[256];                // logits_noise
  __shared__ float sSel[256];               // logits (destroyed by selection)
  __shared__ float sProb[256];              // per-(row,expert) prob
  __shared__ float sImp[16 * KSEL];         // per-(row,slot) score

  const int tid  = threadIdx.x;
  const int wave = tid >> 5;
  const int lane = tid & 31;
  const int hi   = lane >> 4;   // half-wave select
  const int n16  = lane & 15;   // A: row M; B: col N
  const int rowbase = blockIdx.x * 16;

  // ---- Phase 1: WMMA K-loop ------------------------------------------------
  v8f accG = {};
  v8f accN = {};

  const size_t arow = (size_t)(rowbase + n16) * DIM;  // A-matrix row for lane
  const int k0 = wave * KSLICE;
  const int kd = hi * 2;                              // K sub-offset per half

  #pragma unroll 8
  for (int k = k0; k < k0 + KSLICE; k += 4) {
    // A (16x4 f32): lane holds x[M][k+kd], x[M][k+kd+1] -> one b64 NT load
    v2f a = __builtin_nontemporal_load((const v2f*)(x + arow + (size_t)(k + kd)));
    // B (4x16 f32): lane holds w[k+kd][n16], w[k+kd+1][n16] (L2-resident)
    const int kk = (k + kd) * NEXP + n16;
    v2f bg, bn;
    bg.x = wg[kk];        bg.y = wg[kk + NEXP];
    bn.x = wn[kk];        bn.y = wn[kk + NEXP];
    // D = A*B + C  (8 args: neg_a, A, neg_b, B, c_mod, C, reuse_a, reuse_b)
    accG = __builtin_amdgcn_wmma_f32_16x16x4_f32(false, a, false, bg,
                                                 (short)0, accG, false, false);
    accN = __builtin_amdgcn_wmma_f32_16x16x4_f32(false, a, false, bn,
                                                 (short)0, accN, false, false);
  }

  // ---- Phase 2: deterministic cross-wave reduction ------------------------
  #pragma unroll
  for (int v = 0; v < 8; ++v) {
    s_part[((wave * 2 + 0) * 32 + lane) * 8 + v] = accG[v];
    s_part[((wave * 2 + 1) * 32 + lane) * 8 + v] = accN[v];
  }
  __syncthreads();

  if (wave == 0) {
    #pragma unroll
    for (int v = 0; v < 8; ++v) {
      float g = 0.0f, p = 0.0f;
      #pragma unroll
      for (int w = 0; w < NWAVES; ++w) {
        g += s_part[((w * 2 + 0) * 32 + lane) * 8 + v];
        p += s_part[((w * 2 + 1) * 32 + lane) * 8 + v];
      }
      // C/D layout: M = v + 8*hi, N = n16
      const int M = v + 8 * hi;
      sG[M * 16 + n16] = g;
      sZ[M * 16 + n16] = p;
    }
  }
  __syncthreads();

  // ---- Phase 3: per-row epilogue (16 threads, one row each) ---------------
  if (tid < 16) {
    const int r = tid;
    #pragma unroll
    for (int n = 0; n < 16; ++n) {
      const float z  = sZ[r * 16 + n];
      // stable softplus + eps
      const float zc = fmaxf(z, 0.0f) + log1pf(expf(-fabsf(z))) + NOISE_EPS;
      const float ln = noise[(size_t)(rowbase + r) * NEXP + n] * zc;
      sZ [r * 16 + n] = zc;
      sLN[r * 16 + n] = ln;
      sSel[r * 16 + n] = sG[r * 16 + n] + ln;   // total logits
    }

    // top-5 of 16 (selection; first-index wins on ties like lax.top_k)
    float tv[5]; int ti[5];
    #pragma unroll
    for (int i = 0; i < 5; ++i) {
      float best = -INFINITY; int bj = 0;
      #pragma unroll
      for (int j = 0; j < 16; ++j) {
        const float vj = sSel[r * 16 + j];
        if (vj > best) { best = vj; bj = j; }
      }
      tv[i] = best; ti[i] = bj;
      sSel[r * 16 + bj] = -INFINITY;
    }

    // softmax over top-4 (tv[0] is the max since sorted)
    const float e0 = 1.0f;
    const float e1 = expf(tv[1] - tv[0]);
    const float e2 = expf(tv[2] - tv[0]);
    const float e3 = expf(tv[3] - tv[0]);
    const float rden = 1.0f / (e0 + e1 + e2 + e3);
    const float sc[4] = { e0 * rden, e1 * rden, e2 * rden, e3 * rden };

    const size_t rowg = (size_t)(rowbase + r);
    #pragma unroll
    for (int i = 0; i < KSEL; ++i) {
      out[rowg * KSEL + i]                         = (float)ti[i];  // indices
      out[(size_t)N_ROWS * KSEL + rowg * KSEL + i] = sc[i];         // scores
      sImp[r * KSEL + i] = sc[i];
    }

    const float thr_in  = tv[4];  // 5th largest
    const float thr_out = tv[3];  // 4th largest
    const float inv_s2  = 0.70710678118654752f;
    #pragma unroll
    for (int n = 0; n < 16; ++n) {
      const float thr = (sLN[r * 16 + n] > thr_in) ? thr_in : thr_out;
      const float t   = (sG[r * 16 + n] - thr) / sZ[r * 16 + n];
      sProb[r * 16 + n] = 0.5f * erfcf(-t * inv_s2);   // norm.cdf
    }
  }
  __syncthreads();

  // ---- per-block partial sums: load[16] + importance[4] -------------------
  if (tid < 20) {
    float s = 0.0f;
    if (tid < 16) {
      #pragma unroll
      for (int r = 0; r < 16; ++r) s += sProb[r * 16 + tid];
    } else {
      const int i = tid - 16;
      #pragma unroll
      for (int r = 0; r < 16; ++r) s += sImp[r * KSEL + i];
    }
    ws[(size_t)blockIdx.x * 32 + tid] = s;
  }
}

// ---------------------------------------------------------------------------
// Finalize: reduce 512 partials, compute cv_squared for load (n=16) and
// importance (n=4), write gate_loss.
// ---------------------------------------------------------------------------
__global__ __launch_bounds__(32) void gate_finalize_kernel(
    const float* __restrict__ ws, float* __restrict__ out)
{
  __shared__ float s[20];
  const int t = threadIdx.x;
  if (t < 20) {
    float acc = 0.0f;
    for (int b = 0; b < NBLOCKS; ++b) acc += ws[(size_t)b * 32 + t];
    s[t] = acc;
  }
  __syncthreads();
  if (t == 0) {
    // cv_squared(load): 16 values, ddof=1
    float mL = 0.0f;
    for (int i = 0; i < 16; ++i) mL += s[i];
    mL *= (1.0f / 16.0f);
    float vL = 0.0f;
    for (int i = 0; i < 16; ++i) { const float d = s[i] - mL; vL += d * d; }
    vL *= (1.0f / 15.0f);
    const float cvL = vL / (mL * mL + 1e-10f);
    // cv_squared(importance): 4 values, ddof=1
    float mI = 0.0f;
    for (int i = 16; i < 20; ++i) mI += s[i];
    mI *= 0.25f;
    float vI = 0.0f;
    for (int i = 16; i < 20; ++i) { const float d = s[i] - mI; vI += d * d; }
    vI *= (1.0f / 3.0f);
    const float cvI = vI / (mI * mI + 1e-10f);
    out[(size_t)N_ROWS * KSEL * 2] = (cvI + cvL) * LOSS_COEF;
  }
}

extern "C" void kernel_launch(void* const* d_in, const int* in_sizes, int n_in,
                              void* d_out, int out_size, void* d_ws, size_t ws_size,
                              hipStream_t stream) {
  const float* x      = (const float*)d_in[0];
  const float* wgate  = (const float*)d_in[1];
  const float* wnoise = (const float*)d_in[2];
  const float* noise  = (const float*)d_in[3];
  float* out = (float*)d_out;
  float* ws  = (float*)d_ws;

  gate_fused_kernel<<<NBLOCKS, NTHREADS, 0, stream>>>(x, wgate, wnoise, noise, out, ws);
  gate_finalize_kernel<<<1, 32, 0, stream>>>(ws, out);
}